// WideAndDeepModel_5617817223532
// MI455X (gfx1250) — compile-verified
//
#include <hip/hip_runtime.h>

typedef __attribute__((ext_vector_type(2))) float v2f;
typedef __attribute__((ext_vector_type(8))) float v8f;

#define EMB  64
#define MT   32      // batch rows per block
#define P0   132     // LDS pitch for x0 (128 cols) / h2 (128 cols)
#define P1   260     // LDS pitch for h1 (256 cols)
#define P3   68      // LDS pitch for h3 (64 cols)

// eval-mode BN scale: 1/sqrt(1 + 1e-5)
#define BN_SCALE 0.9999950000374997f

// One 16x16 output tile, fp32 WMMA, K accumulated in steps of 4.
// A in LDS row-major [m][k] (pitch lda), B in global row-major [k][n] (pitch ldb).
// ISA VGPR layouts (CDNA5 ISA, 32-bit A 16x4 & mirror for B):
//   A frag: lane L -> m = L&15, k = (L>>4)*2 + {0,1}
//   B frag: lane L -> n = L&15, k = (L>>4)*2 + {0,1}
__device__ __forceinline__ v8f wmma_f32_tile(const float* __restrict__ Alds, int lda,
                                             const float* __restrict__ Bg, int ldb,
                                             int mt, int nt, int K, int lane) {
  const int m  = lane & 15;
  const int n  = lane & 15;
  const int kh = (lane >> 4) << 1;                 // 0 or 2
  const float* arow = Alds + (mt * 16 + m) * lda + kh;
  const float* bcol = Bg + (size_t)kh * ldb + nt * 16 + n;
  v8f acc = {0.f, 0.f, 0.f, 0.f, 0.f, 0.f, 0.f, 0.f};
#pragma unroll 4
  for (int k = 0; k < K; k += 4) {
    v2f a, b;
    a[0] = arow[k];
    a[1] = arow[k + 1];
    b[0] = bcol[(size_t)k * ldb];
    b[1] = bcol[(size_t)(k + 1) * ldb];
    // 8 args: (neg_a, A, neg_b, B, c_mod, C, reuse_a, reuse_b)
    acc = __builtin_amdgcn_wmma_f32_16x16x4_f32(false, a, false, b, (short)0, acc,
                                                false, false);
  }
  return acc;
}

// Fused epilogue: +bias, ReLU, eval-BN (gamma/sqrt(1+eps) * h + beta) -> LDS.
// C/D layout: lane L -> col = L&15, rows = (L>>4)*8 + v for VGPR v.
__device__ __forceinline__ void store_bias_relu_bn(v8f acc, float* __restrict__ Out,
                                                   int ldo, int mt, int nt,
                                                   const float* __restrict__ bias,
                                                   const float* __restrict__ gamma,
                                                   const float* __restrict__ beta,
                                                   int lane) {
  const int nc = nt * 16 + (lane & 15);
  const float bi = bias[nc];
  const float sc = gamma[nc] * BN_SCALE;
  const float sh = beta[nc];
  const int mbase = mt * 16 + ((lane >> 4) << 3);
#pragma unroll
  for (int v = 0; v < 8; ++v) {
    float h = acc[v] + bi;
    h = h > 0.f ? h : 0.f;
    Out[(mbase + v) * ldo + nc] = sc * h + sh;
  }
}

__global__ __launch_bounds__(256) void wide_and_deep_fused(
    const int* __restrict__ user_ids, const int* __restrict__ item_ids,
    const float* __restrict__ wide_w, const float* __restrict__ wide_b,
    const float* __restrict__ user_table, const float* __restrict__ item_table,
    const float* __restrict__ w1, const float* __restrict__ b1,
    const float* __restrict__ g1, const float* __restrict__ be1,
    const float* __restrict__ w2, const float* __restrict__ b2,
    const float* __restrict__ g2, const float* __restrict__ be2,
    const float* __restrict__ w3, const float* __restrict__ b3,
    const float* __restrict__ g3, const float* __restrict__ be3,
    const float* __restrict__ w4, const float* __restrict__ b4,
    float* __restrict__ out, int batch, int num_users) {
  __shared__ float bufA[MT * P0];   // x0 [32x128], later h2 [32x128]
  __shared__ float bufB[MT * P1];   // h1 [32x256], later h3 [32x64] (pitch P3)
  __shared__ float wideBuf[MT];

  const int tid  = threadIdx.x;
  const int lane = tid & 31;
  const int wave = tid >> 5;
  const int row0 = blockIdx.x * MT;

  // ---- Phase 0: gather embeddings -> x0 (LDS), wide path scalar ----
  {
    const int r   = tid >> 3;            // 0..31
    const int seg = tid & 7;             // 8 segs x 16 floats = 128 cols
    int gr = row0 + r;
    if (gr >= batch) gr = batch - 1;
    const int col = seg * 16;
    const float* src =
        (seg < 4) ? user_table + (size_t)user_ids[gr] * EMB + col
                  : item_table + (size_t)item_ids[gr] * EMB + (col - EMB);
    const float4* s4 = (const float4*)src;
    float4* d4 = (float4*)(bufA + r * P0 + col);
    d4[0] = s4[0]; d4[1] = s4[1]; d4[2] = s4[2]; d4[3] = s4[3];
  }
  if (tid < MT) {
    int gr = row0 + tid;
    if (gr >= batch) gr = batch - 1;
    wideBuf[tid] = wide_w[user_ids[gr]] +
                   wide_w[(size_t)num_users + item_ids[gr]] + wide_b[0];
  }
  __syncthreads();

  // ---- Phase 1: h1 = BN(relu(x0 @ w1 + b1))  M=32 N=256 K=128 ----
  // 32 tiles of 16x16; each of 8 waves does 4 (uniform -> EXEC all-ones).
#pragma unroll
  for (int i = 0; i < 4; ++i) {
    const int tile = wave + 8 * i;       // 0..31
    const int nt = tile & 15;
    const int mtb = tile >> 4;
    v8f acc = wmma_f32_tile(bufA, P0, w1, 256, mtb, nt, 128, lane);
    store_bias_relu_bn(acc, bufB, P1, mtb, nt, b1, g1, be1, lane);
  }
  __syncthreads();

  // ---- Phase 2: h2 = BN(relu(h1 @ w2 + b2))  M=32 N=128 K=256 ----
#pragma unroll
  for (int i = 0; i < 2; ++i) {
    const int tile = wave + 8 * i;       // 0..15
    const int nt = tile & 7;
    const int mtb = tile >> 3;
    v8f acc = wmma_f32_tile(bufB, P1, w2, 128, mtb, nt, 256, lane);
    store_bias_relu_bn(acc, bufA, P0, mtb, nt, b2, g2, be2, lane);
  }
  __syncthreads();

  // ---- Phase 3: h3 = BN(relu(h2 @ w3 + b3))  M=32 N=64 K=128 ----
  {
    const int tile = wave;               // 0..7
    const int nt = tile & 3;
    const int mtb = tile >> 2;
    v8f acc = wmma_f32_tile(bufA, P0, w3, 64, mtb, nt, 128, lane);
    store_bias_relu_bn(acc, bufB, P3, mtb, nt, b3, g3, be3, lane);
  }
  __syncthreads();

  // ---- Phase 4: out = wide + h3 @ w4 + b4 ----
  if (tid < MT) {
    const int gr = row0 + tid;
    if (gr < batch) {
      const float* hrow = bufB + tid * P3;
      float s = 0.f;
#pragma unroll
      for (int k = 0; k < EMB; ++k) s += hrow[k] * w4[k];
      out[gr] = wideBuf[tid] + s + b4[0];
    }
  }
}

extern "C" void kernel_launch(void* const* d_in, const int* in_sizes, int n_in,
                              void* d_out, int out_size, void* d_ws, size_t ws_size,
                              hipStream_t stream) {
  const int*   user_ids   = (const int*)d_in[0];
  const int*   item_ids   = (const int*)d_in[1];
  const float* wide_w     = (const float*)d_in[2];
  const float* wide_b     = (const float*)d_in[3];
  const float* user_table = (const float*)d_in[4];
  const float* item_table = (const float*)d_in[5];
  const float* w1  = (const float*)d_in[6];
  const float* b1  = (const float*)d_in[7];
  const float* g1  = (const float*)d_in[8];
  const float* be1 = (const float*)d_in[9];
  const float* w2  = (const float*)d_in[10];
  const float* b2  = (const float*)d_in[11];
  const float* g2  = (const float*)d_in[12];
  const float* be2 = (const float*)d_in[13];
  const float* w3  = (const float*)d_in[14];
  const float* b3  = (const float*)d_in[15];
  const float* g3  = (const float*)d_in[16];
  const float* be3 = (const float*)d_in[17];
  const float* w4  = (const float*)d_in[18];
  const float* b4  = (const float*)d_in[19];

  const int batch     = in_sizes[0];
  const int num_users = in_sizes[4] / EMB;   // user_table rows
  const int blocks    = (batch + MT - 1) / MT;

  wide_and_deep_fused<<<dim3(blocks), dim3(256), 0, stream>>>(
      user_ids, item_ids, wide_w, wide_b, user_table, item_table,
      w1, b1, g1, be1, w2, b2, g2, be2, w3, b3, g3, be3, w4, b4,
      (float*)d_out, batch, num_users);
}